// CGCN_11759620456618
// MI455X (gfx1250) — compile-verified
//
#include <hip/hip_runtime.h>
#include <hip/hip_bf16.h>

typedef float v2f __attribute__((ext_vector_type(2)));
typedef float v8f __attribute__((ext_vector_type(8)));

#define DIMF 128
#define DIMC 64
#define SLOPE 0.01f

// ---------- monotone float <-> uint key for atomic segment-max ----------
__device__ __forceinline__ unsigned fkey(float f) {
  unsigned u = __float_as_uint(f);
  return (u & 0x80000000u) ? ~u : (u | 0x80000000u);
}
__device__ __forceinline__ float funkey(unsigned k) {
  unsigned u = (k & 0x80000000u) ? (k & 0x7FFFFFFFu) : ~k;
  return __uint_as_float(u);
}

// ---------- feats = l2norm(leaky_relu(features @ W^T + b)) via WMMA f32 ----------
// One wave computes a 16(items) x 64(channels) tile, K=128 in steps of 4 using
// V_WMMA_F32_16X16X4_F32.  A layout: lane&15 = row M, lane>=16 holds K pair {2,3}.
// B layout (mirror): lane&15 = col N, same K-pair split.  D: VGPR j = row j / j+8.
// Out-of-range rows load from a clamped (valid) address and are dropped at the
// store, avoiding per-iteration EXEC-mask branches around the A load.
__global__ __launch_bounds__(256) void mlp_feats_kernel(
    const float* __restrict__ A, const float* __restrict__ W,
    const float* __restrict__ bias, float* __restrict__ out, int numItem)
{
  const int lane = threadIdx.x & 31;
  const int wave = threadIdx.x >> 5;
  const int tileBase = (blockIdx.x * 8 + wave) * 16;
  if (tileBase >= numItem) return;              // wave-uniform: EXEC stays all-1s
  const int n0    = lane & 15;
  const int kHalf = (lane >> 4) << 1;           // 0 or 2
  const int mA    = tileBase + n0;
  const int mClamp = mA < numItem ? mA : (numItem - 1);
  const float* arow = A + (size_t)mClamp * DIMF + kHalf;
  const float* w0 = W + (size_t)(n0 +  0) * DIMF + kHalf;
  const float* w1 = W + (size_t)(n0 + 16) * DIMF + kHalf;
  const float* w2 = W + (size_t)(n0 + 32) * DIMF + kHalf;
  const float* w3 = W + (size_t)(n0 + 48) * DIMF + kHalf;

  v8f acc0 = {}, acc1 = {}, acc2 = {}, acc3 = {};
  for (int k = 0; k < DIMF; k += 4) {
    v2f a  = *(const v2f*)(arow + k);
    v2f b0 = *(const v2f*)(w0 + k);
    v2f b1 = *(const v2f*)(w1 + k);
    v2f b2 = *(const v2f*)(w2 + k);
    v2f b3 = *(const v2f*)(w3 + k);
    acc0 = __builtin_amdgcn_wmma_f32_16x16x4_f32(false, a, false, b0, (short)0, acc0, false, false);
    acc1 = __builtin_amdgcn_wmma_f32_16x16x4_f32(false, a, false, b1, (short)0, acc1, false, false);
    acc2 = __builtin_amdgcn_wmma_f32_16x16x4_f32(false, a, false, b2, (short)0, acc2, false, false);
    acc3 = __builtin_amdgcn_wmma_f32_16x16x4_f32(false, a, false, b3, (short)0, acc3, false, false);
  }

  const float bv0 = bias[n0], bv1 = bias[n0 + 16], bv2 = bias[n0 + 32], bv3 = bias[n0 + 48];
  float inv[8];
  #pragma unroll
  for (int j = 0; j < 8; ++j) {
    float v0 = acc0[j] + bv0; v0 = v0 > 0.f ? v0 : SLOPE * v0; acc0[j] = v0;
    float v1 = acc1[j] + bv1; v1 = v1 > 0.f ? v1 : SLOPE * v1; acc1[j] = v1;
    float v2 = acc2[j] + bv2; v2 = v2 > 0.f ? v2 : SLOPE * v2; acc2[j] = v2;
    float v3 = acc3[j] + bv3; v3 = v3 > 0.f ? v3 : SLOPE * v3; acc3[j] = v3;
    float ss = v0*v0 + v1*v1 + v2*v2 + v3*v3;
    ss += __shfl_xor(ss, 1, 32);
    ss += __shfl_xor(ss, 2, 32);
    ss += __shfl_xor(ss, 4, 32);
    ss += __shfl_xor(ss, 8, 32);                 // reduce within each 16-lane half
    inv[j] = 1.0f / fmaxf(sqrtf(ss), 1e-12f);
  }
  #pragma unroll
  for (int j = 0; j < 8; ++j) {
    int row = tileBase + j + ((lane >> 4) << 3);
    if (row < numItem) {
      float* o = out + (size_t)row * DIMC + n0;
      o[0]  = acc0[j] * inv[j];
      o[16] = acc1[j] * inv[j];
      o[32] = acc2[j] * inv[j];
      o[48] = acc3[j] * inv[j];
    }
  }
}

// ---------- out[row] = l2norm(a[row] (+ b[row])) ; 32 lanes / row of 64 ----------
__global__ __launch_bounds__(256) void row_l2norm_kernel(
    const float* __restrict__ a, const float* __restrict__ b,
    float* __restrict__ out, int nRows)
{
  int g = blockIdx.x * 256 + threadIdx.x;
  int row = g >> 5, lane = g & 31;
  if (row >= nRows) return;
  float2 v = ((const float2*)(a + (size_t)row * DIMC))[lane];
  if (b != nullptr) {
    float2 w = ((const float2*)(b + (size_t)row * DIMC))[lane];
    v.x += w.x; v.y += w.y;
  }
  float ss = v.x * v.x + v.y * v.y;
  ss += __shfl_xor(ss, 1, 32);
  ss += __shfl_xor(ss, 2, 32);
  ss += __shfl_xor(ss, 4, 32);
  ss += __shfl_xor(ss, 8, 32);
  ss += __shfl_xor(ss, 16, 32);
  float inv = 1.0f / fmaxf(sqrtf(ss), 1e-12f);
  ((float2*)(out + (size_t)row * DIMC))[lane] = make_float2(v.x * inv, v.y * inv);
}

// ---------- per-edge dot score + atomic segment-max ; 16 lanes / edge ----------
__global__ __launch_bounds__(256) void edge_score_kernel(
    const float* __restrict__ x, const int* __restrict__ srcA, const int* __restrict__ dstA,
    float* __restrict__ score, unsigned* __restrict__ smaxKey, int nE, int bidir)
{
  long long g = (long long)blockIdx.x * 256 + threadIdx.x;
  int e = (int)(g >> 4), sub = (int)(g & 15);
  int total = bidir ? (nE << 1) : nE;
  if (e >= total) return;
  int s, d;
  if (e < nE) { s = srcA[e];      d = dstA[e];      }
  else        { s = dstA[e - nE]; d = srcA[e - nE]; }
  float4 va = ((const float4*)(x + (size_t)s * DIMC))[sub];
  float4 vb = ((const float4*)(x + (size_t)d * DIMC))[sub];
  float p = va.x*vb.x + va.y*vb.y + va.z*vb.z + va.w*vb.w;
  p += __shfl_xor(p, 1, 32);
  p += __shfl_xor(p, 2, 32);
  p += __shfl_xor(p, 4, 32);
  p += __shfl_xor(p, 8, 32);
  if (sub == 0) {
    score[e] = p;
    atomicMax(smaxKey + d, fkey(p));
  }
}

// ---------- ex = exp(score - smax[d]); denom[d] += ex ----------
__global__ __launch_bounds__(256) void edge_exp_kernel(
    float* __restrict__ score, const unsigned* __restrict__ smaxKey,
    float* __restrict__ denom, const int* __restrict__ srcA,
    const int* __restrict__ dstA, int nE, int bidir)
{
  int e = blockIdx.x * 256 + threadIdx.x;
  int total = bidir ? (nE << 1) : nE;
  if (e >= total) return;
  int d = (e < nE) ? dstA[e] : srcA[e - nE];
  float ex = expf(score[e] - funkey(smaxKey[d]));
  score[e] = ex;
  atomicAdd(denom + d, ex);
}

// ---------- xhat[d] += alpha * x[s]; optional alpha output ; 16 lanes / edge ----------
__global__ __launch_bounds__(256) void edge_agg_kernel(
    const float* __restrict__ x, const int* __restrict__ srcA, const int* __restrict__ dstA,
    const float* __restrict__ exv, const float* __restrict__ denom,
    float* __restrict__ xhat, float* __restrict__ alphaOut, int nE, int bidir)
{
  long long g = (long long)blockIdx.x * 256 + threadIdx.x;
  int e = (int)(g >> 4), sub = (int)(g & 15);
  int total = bidir ? (nE << 1) : nE;
  if (e >= total) return;
  int s, d;
  if (e < nE) { s = srcA[e];      d = dstA[e];      }
  else        { s = dstA[e - nE]; d = srcA[e - nE]; }
  float alpha = exv[e] / (denom[d] + 1e-16f);
  float4 va = ((const float4*)(x + (size_t)s * DIMC))[sub];
  float* o = xhat + (size_t)d * DIMC + sub * 4;
  atomicAdd(o + 0, va.x * alpha);
  atomicAdd(o + 1, va.y * alpha);
  atomicAdd(o + 2, va.z * alpha);
  atomicAdd(o + 3, va.w * alpha);
  if (alphaOut != nullptr && sub == 0) alphaOut[e] = alpha;
}

// ---------- io = x + leaky_relu(io)   (io holds x_hat, becomes final output) ----------
__global__ __launch_bounds__(256) void add_act_kernel(
    const float* __restrict__ x, float* __restrict__ io, long long n)
{
  long long i = (long long)blockIdx.x * 256 + threadIdx.x;
  if (i >= n) return;
  float v = io[i];
  v = v > 0.f ? v : SLOPE * v;
  io[i] = x[i] + v;
}

extern "C" void kernel_launch(void* const* d_in, const int* in_sizes, int n_in,
                              void* d_out, int out_size, void* d_ws, size_t ws_size,
                              hipStream_t stream)
{
  (void)n_in; (void)out_size; (void)ws_size;
  const float* features   = (const float*)d_in[0];
  const float* preference = (const float*)d_in[1];
  const float* mlp_w      = (const float*)d_in[2];
  const float* mlp_b      = (const float*)d_in[3];
  const int*   edge       = (const int*)d_in[4];

  const int numItem  = in_sizes[0] / DIMF;   // 100000
  const int numUser  = in_sizes[1] / DIMC;   // 200000
  const int nE       = in_sizes[4] / 2;      // 1000000
  const int numNodes = numUser + numItem;    // 300000
  const int* srcA = edge;
  const int* dstA = edge + nE;

  // workspace: x | smaxKey | denom | exbuf   (~87 MB)
  float*    x       = (float*)d_ws;                                  // numNodes*64
  unsigned* smaxKey = (unsigned*)(x + (size_t)numNodes * DIMC);      // numNodes
  float*    denom   = (float*)(smaxKey + numNodes);                  // numNodes
  float*    exbuf   = denom + numNodes;                              // 2*nE

  float* outX     = (float*)d_out;                    // node output region (also xhat scratch)
  float* outAlpha = outX + (size_t)numNodes * DIMC;   // per-edge alpha region
  float* xhat     = outX;                             // accumulate x_hat here, finalize in place

  // 1) feats via WMMA GEMM + bias + leaky_relu + row-l2norm -> x[numUser..]
  {
    int tiles  = (numItem + 15) / 16;
    int blocks = (tiles + 7) / 8;
    mlp_feats_kernel<<<blocks, 256, 0, stream>>>(features, mlp_w, mlp_b,
                                                 x + (size_t)numUser * DIMC, numItem);
  }
  // 2) pref = l2norm(preference) -> x[0..numUser)
  row_l2norm_kernel<<<(numUser * 32 + 255) / 256, 256, 0, stream>>>(preference, nullptr, x, numUser);

  // 3) routing iterations (dst < numUser by construction)
  for (int it = 0; it < 3; ++it) {
    hipMemsetAsync(smaxKey, 0, (size_t)numUser * sizeof(unsigned), stream);
    hipMemsetAsync(denom,   0, (size_t)numUser * sizeof(float), stream);
    hipMemsetAsync(xhat,    0, (size_t)numUser * DIMC * sizeof(float), stream);
    {
      long long t = (long long)nE * 16;
      edge_score_kernel<<<(unsigned)((t + 255) / 256), 256, 0, stream>>>(x, srcA, dstA, exbuf, smaxKey, nE, 0);
    }
    edge_exp_kernel<<<(nE + 255) / 256, 256, 0, stream>>>(exbuf, smaxKey, denom, srcA, dstA, nE, 0);
    {
      long long t = (long long)nE * 16;
      edge_agg_kernel<<<(unsigned)((t + 255) / 256), 256, 0, stream>>>(x, srcA, dstA, exbuf, denom, xhat, nullptr, nE, 0);
    }
    // pref = l2norm(pref + xhat[:numUser]) in place in x
    row_l2norm_kernel<<<(numUser * 32 + 255) / 256, 256, 0, stream>>>(x, xhat, x, numUser);
  }

  // 4) final bidirectional pass over all nodes
  hipMemsetAsync(smaxKey, 0, (size_t)numNodes * sizeof(unsigned), stream);
  hipMemsetAsync(denom,   0, (size_t)numNodes * sizeof(float), stream);
  hipMemsetAsync(xhat,    0, (size_t)numNodes * DIMC * sizeof(float), stream);
  {
    long long t = (long long)nE * 32;
    edge_score_kernel<<<(unsigned)((t + 255) / 256), 256, 0, stream>>>(x, srcA, dstA, exbuf, smaxKey, nE, 1);
    edge_exp_kernel<<<(nE * 2 + 255) / 256, 256, 0, stream>>>(exbuf, smaxKey, denom, srcA, dstA, nE, 1);
    edge_agg_kernel<<<(unsigned)((t + 255) / 256), 256, 0, stream>>>(x, srcA, dstA, exbuf, denom, xhat, outAlpha, nE, 1);
  }
  // out = x + leaky_relu(x_hat)  (finalizes d_out node region in place)
  {
    long long n = (long long)numNodes * DIMC;
    add_act_kernel<<<(unsigned)((n + 255) / 256), 256, 0, stream>>>(x, outX, n);
  }
}